// SparseAutoencoder_64029372449140
// MI455X (gfx1250) — compile-verified
//
#include <hip/hip_runtime.h>
#include <hip/hip_bf16.h>
#include <stdint.h>

typedef float v2f  __attribute__((ext_vector_type(2)));
typedef float v8f  __attribute__((ext_vector_type(8)));
typedef float f32x4 __attribute__((ext_vector_type(4)));

#define BATCH 8192
#define INSZ  768
#define HID   16384

// GEMM tiling
#define BM 128
#define BN 128
#define BK 16
#define LDA 20   // padded LDS row stride (floats); keeps 16B align, avoids bank conflicts

// ---------------------------------------------------------------------------
// CDNA5 async copy helpers: global -> LDS without a VGPR round-trip.
// Tracked by ASYNCcnt (s_wait_asynccnt), per CDNA5 ISA.
// Generic LDS pointer low 32 bits == LDS byte offset (flat aperture rule).
// ---------------------------------------------------------------------------
__device__ __forceinline__ void async_b128(unsigned lds_off, const float* g) {
  asm volatile("global_load_async_to_lds_b128 %0, %1, off"
               :: "v"(lds_off), "v"(g) : "memory");
}
__device__ __forceinline__ void wait_async0() {
#if __has_builtin(__builtin_amdgcn_s_wait_asynccnt)
  __builtin_amdgcn_s_wait_asynccnt(0);
#else
  asm volatile("s_wait_asynccnt 0x0" ::: "memory");
#endif
}

// ---------------------------------------------------------------------------
// Kernel 1: pre = x @ W^T + b  via V_WMMA_F32_16X16X4_F32
// Grid: (HID/BN, BATCH/BM), Block: 256 (8 waves; wave grid 2(M) x 4(N), 64x32 per wave)
// Double-buffered LDS, async global->LDS copies overlapping the WMMA stream.
// ---------------------------------------------------------------------------
__global__ __launch_bounds__(256) void sae_encoder_gemm(
    const float* __restrict__ x, const float* __restrict__ W,
    const float* __restrict__ bias, float* __restrict__ pre)
{
  __shared__ float As[2][BM * LDA];
  __shared__ float Bs[2][BN * LDA];

  const int t     = threadIdx.x;
  const int lane  = t & 31;
  const int wave  = t >> 5;
  const int waveM = wave & 1;   // 0..1 -> 64-row strip
  const int waveN = wave >> 1;  // 0..3 -> 32-col strip
  const int m0 = blockIdx.y * BM;
  const int n0 = blockIdx.x * BN;

  const int lhalf = lane >> 4;  // 0: K={0,1}, 1: K={2,3} (f32 A/B lane layout)
  const int l15   = lane & 15;

  // Per-thread staging coords: 2 float4 per matrix per chunk (512 float4 / 256 thr)
  const int f4a = t * 2, f4b = t * 2 + 1;
  const int r0 = f4a >> 2, c0 = (f4a & 3) << 2;
  const int r1 = f4b >> 2, c1 = (f4b & 3) << 2;
  const float* gA0 = x + (size_t)(m0 + r0) * INSZ + c0;
  const float* gA1 = x + (size_t)(m0 + r1) * INSZ + c1;
  const float* gB0 = W + (size_t)(n0 + r0) * INSZ + c0;
  const float* gB1 = W + (size_t)(n0 + r1) * INSZ + c1;
  unsigned oA0[2], oA1[2], oB0[2], oB1[2];
  #pragma unroll
  for (int b = 0; b < 2; ++b) {
    oA0[b] = (unsigned)(uintptr_t)&As[b][r0 * LDA + c0];
    oA1[b] = (unsigned)(uintptr_t)&As[b][r1 * LDA + c1];
    oB0[b] = (unsigned)(uintptr_t)&Bs[b][r0 * LDA + c0];
    oB1[b] = (unsigned)(uintptr_t)&Bs[b][r1 * LDA + c1];
  }

  v8f acc[4][2];
  #pragma unroll
  for (int mi = 0; mi < 4; ++mi)
    #pragma unroll
    for (int ni = 0; ni < 2; ++ni)
      acc[mi][ni] = {};

  // Prologue: stage chunk 0 into buffer 0
  async_b128(oA0[0], gA0);
  async_b128(oA1[0], gA1);
  async_b128(oB0[0], gB0);
  async_b128(oB1[0], gB1);

  const int NCH = INSZ / BK;   // 48 chunks
  for (int i = 0; i < NCH; ++i) {
    const int cur = i & 1;
    wait_async0();         // this wave's chunk-i transfers are in LDS
    __syncthreads();       // everyone's chunk-i data present; buf[1-cur] free

    if (i + 1 < NCH) {     // overlap chunk i+1 copies with chunk i compute
      const int nxt = 1 - cur;
      const int ko = (i + 1) * BK;
      async_b128(oA0[nxt], gA0 + ko);
      async_b128(oA1[nxt], gA1 + ko);
      async_b128(oB0[nxt], gB0 + ko);
      async_b128(oB1[nxt], gB1 + ko);
    }

    const float* Ac = &As[cur][0];
    const float* Bc = &Bs[cur][0];
    #pragma unroll
    for (int kk = 0; kk < BK; kk += 4) {
      v2f af[4], bf[2];
      #pragma unroll
      for (int mi = 0; mi < 4; ++mi) {
        int row = waveM * 64 + mi * 16 + l15;          // A: 16x4, lane holds K pair
        af[mi] = *(const v2f*)(&Ac[row * LDA + kk + 2 * lhalf]);
      }
      #pragma unroll
      for (int ni = 0; ni < 2; ++ni) {
        int col = waveN * 32 + ni * 16 + l15;          // B: 4x16, N across lanes
        bf[ni] = *(const v2f*)(&Bc[col * LDA + kk + 2 * lhalf]);
      }
      #pragma unroll
      for (int mi = 0; mi < 4; ++mi)
        #pragma unroll
        for (int ni = 0; ni < 2; ++ni)
          acc[mi][ni] = __builtin_amdgcn_wmma_f32_16x16x4_f32(
              false, af[mi], false, bf[ni], (short)0, acc[mi][ni], false, false);
    }
  }

  // Epilogue: add bias, write pre into the encoded slice of d_out
  #pragma unroll
  for (int mi = 0; mi < 4; ++mi)
    #pragma unroll
    for (int ni = 0; ni < 2; ++ni) {
      int n = n0 + waveN * 32 + ni * 16 + l15;
      float bn = bias[n];
      #pragma unroll
      for (int r = 0; r < 8; ++r) {
        int m = m0 + waveM * 64 + mi * 16 + lhalf * 8 + r;  // C layout: VGPR r -> M=r(+8)
        pre[(size_t)m * HID + n] = acc[mi][ni][r] + bn;
      }
    }
}

// ---------------------------------------------------------------------------
// Kernel 2: invn[h] = 1/||W[h,:]||   (into d_ws, 64KB)
// ---------------------------------------------------------------------------
__global__ __launch_bounds__(256) void sae_row_invnorm(
    const float* __restrict__ W, float* __restrict__ invn)
{
  __shared__ float red[256];
  const int h = blockIdx.x;
  const int t = threadIdx.x;
  const float* r = W + (size_t)h * INSZ;
  float s = 0.f;
  #pragma unroll
  for (int j = 0; j < 3; ++j) { float w = r[t + 256 * j]; s += w * w; }
  red[t] = s;
  __syncthreads();
  for (int st = 128; st > 0; st >>= 1) {
    if (t < st) red[t] += red[t + st];
    __syncthreads();
  }
  if (t == 0) invn[h] = rsqrtf(red[0]);
}

// ---------------------------------------------------------------------------
// Kernel 3: exact k-th largest per row via bitwise binary search on
// order-preserving u32 keys; mask encoded in place (pre >= thresh).
// ---------------------------------------------------------------------------
__global__ __launch_bounds__(256) void sae_topk_mask(
    float* __restrict__ enc, const int* __restrict__ kptr)
{
  __shared__ int red[256];
  const int row = blockIdx.x;
  const int t   = threadIdx.x;
  const int k   = *kptr;
  float* p = enc + (size_t)row * HID;

  float    v[64];
  unsigned key[64];
  #pragma unroll
  for (int j = 0; j < 64; ++j) {
    float f = p[t + 256 * j];
    v[j] = f;
    unsigned b = __float_as_uint(f);
    key[j] = (b & 0x80000000u) ? ~b : (b | 0x80000000u);  // monotone f32 -> u32
  }

  unsigned long long L = 0, R = 0xFFFFFFFFull;
  while (L < R) {
    unsigned long long mid = (L + R + 1) >> 1;
    unsigned m32 = (unsigned)mid;
    int c = 0;
    #pragma unroll
    for (int j = 0; j < 64; ++j) c += (key[j] >= m32) ? 1 : 0;
    red[t] = c;
    __syncthreads();
    for (int s = 128; s > 0; s >>= 1) {
      if (t < s) red[t] += red[t + s];
      __syncthreads();
    }
    int total = red[0];
    __syncthreads();
    if (total >= k) L = mid; else R = mid - 1;
  }
  unsigned thr = (unsigned)L;   // key of k-th largest value

  #pragma unroll
  for (int j = 0; j < 64; ++j)
    p[t + 256 * j] = (key[j] >= thr) ? v[j] : 0.0f;
}

// ---------------------------------------------------------------------------
// Kernel 4: decoded[row,:] = sum_{h active} enc[row,h]*invn[h]*W[h,:]
// Deterministic compaction (prefix scan, no atomics) -> identical fp order.
// ---------------------------------------------------------------------------
#define MAXA 2048
__global__ __launch_bounds__(256) void sae_decode(
    const float* __restrict__ enc, const float* __restrict__ W,
    const float* __restrict__ invn, float* __restrict__ dec)
{
  __shared__ int   scan[256];
  __shared__ int   sidx[MAXA];
  __shared__ float sval[MAXA];
  const int row = blockIdx.x;
  const int t   = threadIdx.x;
  const float* p = enc + (size_t)row * HID;

  float v[64];
  int c = 0;
  #pragma unroll
  for (int j = 0; j < 64; ++j) {
    v[j] = p[t + 256 * j];               // coalesced
    c += (v[j] != 0.0f) ? 1 : 0;
  }

  scan[t] = c;
  __syncthreads();
  for (int off = 1; off < 256; off <<= 1) {   // Hillis-Steele inclusive scan
    int val = scan[t];
    int add = (t >= off) ? scan[t - off] : 0;
    __syncthreads();
    scan[t] = val + add;
    __syncthreads();
  }
  int pos   = scan[t] - c;       // exclusive offset for this thread
  int total = scan[255];
  __syncthreads();

  #pragma unroll
  for (int j = 0; j < 64; ++j) {
    if (v[j] != 0.0f && pos < MAXA) {
      int h = t + 256 * j;
      sidx[pos] = h;
      sval[pos] = v[j] * invn[h];
      ++pos;
    }
  }
  __syncthreads();

  int n = total < MAXA ? total : MAXA;   // expected ~k=32
  #pragma unroll
  for (int ii = 0; ii < 3; ++ii) {
    int i = t + 256 * ii;                // 768 = 3*256 output columns
    float a = 0.f;
    for (int j = 0; j < n; ++j)
      a += sval[j] * W[(size_t)sidx[j] * INSZ + i];
    dec[(size_t)row * INSZ + i] = a;
  }
}

// ---------------------------------------------------------------------------
extern "C" void kernel_launch(void* const* d_in, const int* in_sizes, int n_in,
                              void* d_out, int out_size, void* d_ws, size_t ws_size,
                              hipStream_t stream) {
  const float* x    = (const float*)d_in[0];   // (8192, 768)
  const float* W    = (const float*)d_in[1];   // (16384, 768)
  const float* bias = (const float*)d_in[2];   // (16384,)
  const int*   kptr = (const int*)d_in[3];     // scalar k

  float* out  = (float*)d_out;
  float* dec  = out;                                   // (8192, 768)
  float* enc  = out + (size_t)BATCH * INSZ;            // (8192, 16384)
  float* invn = (float*)d_ws;                          // 16384 floats

  dim3 gGemm(HID / BN, BATCH / BM);  // (128, 64)
  sae_encoder_gemm<<<gGemm, 256, 0, stream>>>(x, W, bias, enc);
  sae_row_invnorm<<<HID, 256, 0, stream>>>(W, invn);
  sae_topk_mask<<<BATCH, 256, 0, stream>>>(enc, kptr);
  sae_decode<<<BATCH, 256, 0, stream>>>(enc, W, invn, dec);
}